// QwenSentenceEncoder_3977139716345
// MI455X (gfx1250) — compile-verified
//
#include <hip/hip_runtime.h>
#include <hip/hip_bf16.h>

// Problem constants (match reference): B=16, L=4096, D=1024, K=64
#define BN 16
#define LN 4096
#define DN 1024
#define KN 64

typedef __attribute__((ext_vector_type(2))) float v2f;
typedef __attribute__((ext_vector_type(8))) float v8f;

// ---------------------------------------------------------------------------
// Kernel 1: per-span bookkeeping.
// One wave per (b,k) span: clamped [start,end), mask_sum (shfl_xor reduce),
// scale = valid ? 1/denom_safe : 0 (folds masked/plain-mean select + valid
// zeroing).  Also writes sent_mask.
// ---------------------------------------------------------------------------
__global__ __launch_bounds__(256) void span_info_kernel(
    const int* __restrict__ am, const long long* __restrict__ spans,
    int* __restrict__ wsStart, int* __restrict__ wsEnd,
    float* __restrict__ wsUseAm, float* __restrict__ wsScale,
    float* __restrict__ sentMask)
{
    const int wave = threadIdx.x >> 5;
    const int lane = threadIdx.x & 31;
    const int idx  = blockIdx.x * 8 + wave;          // span index in [0, B*K)
    if (idx >= BN * KN) return;
    const int b = idx / KN;

    long long s64 = spans[(size_t)idx * 2 + 0];
    long long e64 = spans[(size_t)idx * 2 + 1];
    int s = (int)(s64 < (long long)LN ? s64 : (long long)LN);
    int e = (int)(e64 < (long long)LN ? e64 : (long long)LN);
    if (s < 0) s = 0;
    if (e < 0) e = 0;
    const int len = e - s;

    int cnt = 0;
    const int* amb = am + (size_t)b * LN;
    for (int l = s + lane; l < e; l += 32) cnt += amb[l];
    #pragma unroll
    for (int off = 16; off > 0; off >>= 1) cnt += __shfl_xor(cnt, off, 32);

    if (lane == 0) {
        const bool useAm = cnt > 0;
        const bool valid = len > 0;
        const float denom = useAm ? (float)cnt : (float)len;
        const float scale = valid ? (1.0f / denom) : 0.0f;
        wsStart[idx]  = s;
        wsEnd[idx]    = e;
        wsUseAm[idx]  = useAm ? 1.0f : 0.0f;
        wsScale[idx]  = scale;
        sentMask[idx] = valid ? 1.0f : 0.0f;
    }
}

// ---------------------------------------------------------------------------
// One K-step of 4 tokens starting at l0 (multiple of 4, caller guarantees
// l0+3 < L): build A from span weights, load B from X, accumulate via
// V_WMMA_F32_16X16X4_F32.
//
// Layouts (ISA 7.12.2), h = lane/16, m = lane%16:
//   A (16x4):  a[0]=(M=m,K=2h)  a[1]=(M=m,K=2h+1)
//   B (4x16):  b[0]=(K=2h,N=m)  b[1]=(K=2h+1,N=m)
//   C (16x16): c[j]=(M=j+8h,N=m)
// A and B want the same token pair (l0+2h, l0+2h+1) per lane.
// ---------------------------------------------------------------------------
__device__ __forceinline__ void wmma_step(
    const float* __restrict__ Xcol, const int* __restrict__ amb,
    int l0, int h, int sm, int em, bool useAm, v8f& c)
{
    const int la = l0 + 2 * h;               // even; la+1 <= L-1: no guards
    const int2 amp = *(const int2*)(amb + la);
    const float x0 = Xcol[(size_t)la * DN];
    const float x1 = Xcol[(size_t)(la + 1) * DN];
    const float w0 = (la     >= sm && la     < em) ? (useAm ? (float)amp.x : 1.0f) : 0.0f;
    const float w1 = (la + 1 >= sm && la + 1 < em) ? (useAm ? (float)amp.y : 1.0f) : 0.0f;
    v2f a  = {w0, w1};
    v2f bb = {x0, x1};
    c = __builtin_amdgcn_wmma_f32_16x16x4_f32(
            false, a, false, bb, (short)0, c, false, false);
}

// ---------------------------------------------------------------------------
// Kernel 2: span pooling as f32 WMMA.  One wave owns a 16-span x 16-channel
// C tile; marches the union token interval of its 16 spans, 16 tokens per
// iteration (4 WMMA steps, immediate-offset load clause) into two alternating
// accumulators (pipelines loads, breaks the WMMA C->C dependency chain).
// ---------------------------------------------------------------------------
__global__ __launch_bounds__(128) void pool_wmma_kernel(
    const float* __restrict__ X, const int* __restrict__ am,
    const int* __restrict__ wsStart, const int* __restrict__ wsEnd,
    const float* __restrict__ wsUseAm, const float* __restrict__ wsScale,
    float* __restrict__ H)
{
    const int lane = threadIdx.x & 31;
    const int wave = threadIdx.x >> 5;
    const int b    = blockIdx.x >> 2;     // 4 m-tiles (of 16 spans) per batch
    const int mt   = blockIdx.x & 3;
    const int k0   = mt * 16;
    const int h    = lane >> 4;
    const int m    = lane & 15;
    const int col  = (blockIdx.y * 4 + wave) * 16 + m;   // channel index

    const int base  = b * KN + k0;
    const int   sm  = wsStart[base + m];
    const int   em  = wsEnd[base + m];
    const bool  useAm = wsUseAm[base + m] != 0.0f;
    const float scl   = wsScale[base + m];

    // Union of the 16 spans via shfl_xor butterfly over per-lane (sm,em),
    // then force scalar so loop control is SALU and EXEC stays all-ones.
    int minS = (em > sm) ? sm : LN;
    int maxE = (em > sm) ? em : 0;
    #pragma unroll
    for (int off = 16; off > 0; off >>= 1) {
        minS = min(minS, __shfl_xor(minS, off, 32));
        maxE = max(maxE, __shfl_xor(maxE, off, 32));
    }
    minS = __builtin_amdgcn_readfirstlane(minS);
    maxE = __builtin_amdgcn_readfirstlane(maxE);

    v8f c0 = {0.f, 0.f, 0.f, 0.f, 0.f, 0.f, 0.f, 0.f};
    v8f c1 = {0.f, 0.f, 0.f, 0.f, 0.f, 0.f, 0.f, 0.f};
    const float* Xcol = X + (size_t)b * LN * DN + col;
    const int*   amb  = am + (size_t)b * LN;

    int l0 = minS & ~3;
    // NOTE: maxE <= L and l0 % 4 == 0  =>  every token touched is < L.
    const int span  = (maxE - l0) > 0 ? (maxE - l0) : 0;
    const int lEnd16 = l0 + (span & ~15);
    for (; l0 < lEnd16; l0 += 16) {
        // Prefetch 64 tokens ahead (gfx1250 global_prefetch_b8).
        int lp = l0 + 64; lp = (lp < LN) ? lp : (LN - 1);
        __builtin_prefetch(Xcol + (size_t)lp * DN, 0, 3);
        // 4 steps, fixed immediate offsets off the same base -> one big
        // load clause; accumulators alternate to avoid D->C chaining.
        wmma_step(Xcol, amb, l0,      h, sm, em, useAm, c0);
        wmma_step(Xcol, amb, l0 + 4,  h, sm, em, useAm, c1);
        wmma_step(Xcol, amb, l0 + 8,  h, sm, em, useAm, c0);
        wmma_step(Xcol, amb, l0 + 12, h, sm, em, useAm, c1);
    }
    for (; l0 < maxE; l0 += 4) {
        wmma_step(Xcol, amb, l0, h, sm, em, useAm, c0);
    }

    // Merge accumulators, broadcast per-row scale (lanes 0..15 hold m=0..15),
    // store final H.
    #pragma unroll
    for (int j = 0; j < 8; ++j) {
        const int row = j + 8 * h;
        const float s = __shfl(scl, row, 32);
        H[(size_t)(base + row) * DN + col] = (c0[j] + c1[j]) * s;
    }
}

// ---------------------------------------------------------------------------
// Launch: inputs [token_hidden f32, attention_mask i32, sent_spans i64].
// Output: H [B,K,D] f32 followed by sent_mask [B,K] f32.
// ---------------------------------------------------------------------------
extern "C" void kernel_launch(void* const* d_in, const int* in_sizes, int n_in,
                              void* d_out, int out_size, void* d_ws, size_t ws_size,
                              hipStream_t stream) {
    const float*     X     = (const float*)d_in[0];
    const int*       am    = (const int*)d_in[1];
    const long long* spans = (const long long*)d_in[2];

    float* out      = (float*)d_out;
    float* H        = out;
    float* sentMask = out + (size_t)BN * KN * DN;

    int*   wsStart = (int*)d_ws;
    int*   wsEnd   = wsStart + BN * KN;
    float* wsUseAm = (float*)(wsEnd + BN * KN);
    float* wsScale = wsUseAm + BN * KN;

    span_info_kernel<<<dim3((BN * KN) / 8), dim3(256), 0, stream>>>(
        am, spans, wsStart, wsEnd, wsUseAm, wsScale, sentMask);

    pool_wmma_kernel<<<dim3(BN * 4, DN / 64), dim3(128), 0, stream>>>(
        X, am, wsStart, wsEnd, wsUseAm, wsScale, H);
}